// Involution_27633819583088
// MI455X (gfx1250) — compile-verified
//
#include <hip/hip_runtime.h>

// ---- problem constants (match reference) ----
#define C_IN   256
#define CR      64
#define G_NUM   16
#define KS       7
#define KK      49          // KS*KS
#define OUT2   784          // KK * G_NUM
#define H_DIM   56
#define W_DIM   56
#define HW    3136
#define B_TOT    8
#define EPSV  1e-5f

#define WAVES    4          // waves per block in the WMMA kernel
#define TILE_N  16          // pixels per wave tile
#define TILES_PER_BATCH (HW / TILE_N)   // 196

typedef float v2f __attribute__((ext_vector_type(2)));
typedef float v8f __attribute__((ext_vector_type(8)));

// =====================================================================
// Kernel A: weight-generation branch, fully fused:
//   wgt(784 x HW) = W2 * relu( bn( W1 * X + b1 ) ) + b2   (per batch)
// Each wave computes a 16-pixel column tile with V_WMMA_F32_16X16X4_F32.
// VGPR layouts per CDNA5 ISA 7.12.2:
//   A 16x4 : lane = M + 16*(K>=2), vgpr = K&1
//   B 4x16 : lane = N + 16*(K>=2), vgpr = K&1
//   C 16x16: vgpr r -> M = r + 8*(lane>=16), N = lane&15
// =====================================================================
__global__ __launch_bounds__(WAVES * 32) void invol_wgt_kernel(
    const float* __restrict__ x,  const float* __restrict__ W1,
    const float* __restrict__ b1, const float* __restrict__ gamma,
    const float* __restrict__ beta, const float* __restrict__ mean,
    const float* __restrict__ var, const float* __restrict__ W2,
    const float* __restrict__ b2, float* __restrict__ wgt,
    int b_start, int nb)
{
    __shared__ float scale_s[CR];
    __shared__ float shift_s[CR];
    __shared__ float b2_s[OUT2];
    __shared__ float y_s[WAVES * CR * TILE_N];   // per-wave y staging (64x16)

    const int tid = threadIdx.x;
    // fold conv1-bias + BN(eval) into scale/shift:  y = acc*scale + shift
    if (tid < CR) {
        float sc = gamma[tid] * rsqrtf(var[tid] + EPSV);
        scale_s[tid] = sc;
        shift_s[tid] = (b1[tid] - mean[tid]) * sc + beta[tid];
    }
    for (int i = tid; i < OUT2; i += WAVES * 32) b2_s[i] = b2[i];
    __syncthreads();

    const int wave = tid >> 5;
    const int lane = tid & 31;
    const int col  = lane & 15;      // N index (pixel within tile)
    const int half = lane >> 4;      // upper 16 lanes -> K/M offset
    const int h2   = half * 2;       // K offset for A/B operands

    const int t = blockIdx.x * WAVES + wave;          // wave-uniform
    if (t >= nb * TILES_PER_BATCH) return;            // uniform exit (EXEC stays all-ones)
    const int b      = b_start + t / TILES_PER_BATCH;
    const int tloc   = t % TILES_PER_BATCH;
    const int n_base = tloc * TILE_N;

    // ---------------- conv1: y(64x16) = W1(64x256) x X(256x16) ----------------
    const float* xcol = x + (size_t)b * C_IN * HW + n_base + col;
    v8f vz = {};
    v8f acc[4];
    acc[0] = vz; acc[1] = vz; acc[2] = vz; acc[3] = vz;

    #pragma unroll 4
    for (int kc = 0; kc < C_IN; kc += 4) {
        v2f bv;                                   // B[k][n] = X[kc+h2+v][col]
        bv.x = xcol[(kc + h2)     * HW];
        bv.y = xcol[(kc + h2 + 1) * HW];
        #pragma unroll
        for (int mt = 0; mt < 4; ++mt) {          // A[m][k] = W1[mt*16+col][kc+h2+v]
            const float* ap = W1 + (mt * 16 + col) * C_IN + kc + h2;
            v2f av; av.x = ap[0]; av.y = ap[1];
            acc[mt] = __builtin_amdgcn_wmma_f32_16x16x4_f32(
                false, av, false, bv, (short)0, acc[mt], false, false);
        }
    }

    // ---------------- BN + ReLU, stage y into LDS ----------------
    float* yw = y_s + wave * (CR * TILE_N);
    #pragma unroll
    for (int mt = 0; mt < 4; ++mt) {
        #pragma unroll
        for (int r = 0; r < 8; ++r) {
            int   o = mt * 16 + r + half * 8;     // C/D row mapping
            float v = acc[mt][r] * scale_s[o] + shift_s[o];
            yw[o * TILE_N + col] = fmaxf(v, 0.0f);
        }
    }

    // ---------------- conv2: wgt(784x16) = W2(784x64) x y(64x16) ----------------
    float* wb = wgt + (size_t)(b - b_start) * OUT2 * HW + n_base + col;
    for (int mt2 = 0; mt2 < KK; ++mt2) {          // 49 tiles of 16 output rows
        v8f d = vz;
        #pragma unroll 4
        for (int kc = 0; kc < CR; kc += 4) {
            v2f bv;                               // B from LDS y
            bv.x = yw[(kc + h2)     * TILE_N + col];
            bv.y = yw[(kc + h2 + 1) * TILE_N + col];
            const float* ap = W2 + (mt2 * 16 + col) * CR + kc + h2;
            v2f av; av.x = ap[0]; av.y = ap[1];
            d = __builtin_amdgcn_wmma_f32_16x16x4_f32(
                false, av, false, bv, (short)0, d, false, false);
        }
        #pragma unroll
        for (int r = 0; r < 8; ++r) {
            int o2 = mt2 * 16 + r + half * 8;
            wb[(size_t)o2 * HW] = d[r] + b2_s[o2];
        }
    }
}

// =====================================================================
// Kernel B: apply per-pixel involution weights.
//   out[b,c,h,w] = sum_k x[b,c,h+ki-3,w+kj-3] * wgt[b, c/16, k, h, w]
// wgt was just written (per-batch, 9.8 MB) -> L2-resident reads.
// =====================================================================
__global__ __launch_bounds__(256) void invol_apply_kernel(
    const float* __restrict__ x, const float* __restrict__ wgt,
    float* __restrict__ out, int b_start, int nb)
{
    int idx  = blockIdx.x * 256 + threadIdx.x;
    int bloc = idx / (C_IN * HW);
    int rem  = idx - bloc * (C_IN * HW);
    int c    = rem / HW;
    int p    = rem - c * HW;
    int h    = p / W_DIM;
    int w    = p - h * W_DIM;
    int b    = b_start + bloc;

    const float* xp = x   + ((size_t)b * C_IN + c) * HW;
    const float* wp = wgt + ((size_t)bloc * OUT2 + (c >> 4) * KK) * HW + p;

    float acc = 0.0f;
    #pragma unroll
    for (int ki = 0; ki < KS; ++ki) {
        int hh = h + ki - 3;
        if ((unsigned)hh < (unsigned)H_DIM) {
            #pragma unroll
            for (int kj = 0; kj < KS; ++kj) {
                int ww = w + kj - 3;
                if ((unsigned)ww < (unsigned)W_DIM) {
                    acc = fmaf(xp[hh * W_DIM + ww], wp[(ki * KS + kj) * HW], acc);
                }
            }
        }
    }
    out[((size_t)b * C_IN + c) * HW + p] = acc;
}

// =====================================================================
extern "C" void kernel_launch(void* const* d_in, const int* in_sizes, int n_in,
                              void* d_out, int out_size, void* d_ws, size_t ws_size,
                              hipStream_t stream)
{
    (void)in_sizes; (void)n_in; (void)out_size;
    const float* x     = (const float*)d_in[0];
    const float* W1    = (const float*)d_in[1];
    const float* b1    = (const float*)d_in[2];
    const float* gamma = (const float*)d_in[3];
    const float* beta  = (const float*)d_in[4];
    const float* mean  = (const float*)d_in[5];
    const float* var   = (const float*)d_in[6];
    const float* W2    = (const float*)d_in[7];
    const float* b2    = (const float*)d_in[8];
    float* out = (float*)d_out;
    float* wgt = (float*)d_ws;

    const size_t per_batch = (size_t)OUT2 * HW * sizeof(float);  // ~9.8 MB
    int nb_stage = (int)(ws_size / per_batch);
    if (nb_stage < 1) nb_stage = 1;
    if (nb_stage > B_TOT) nb_stage = B_TOT;

    for (int b = 0; b < B_TOT; b += nb_stage) {
        int nb = (B_TOT - b < nb_stage) ? (B_TOT - b) : nb_stage;
        dim3 gA(nb * (TILES_PER_BATCH / WAVES));   // nb * 49 blocks of 128
        invol_wgt_kernel<<<gA, WAVES * 32, 0, stream>>>(
            x, W1, b1, gamma, beta, mean, var, W2, b2, wgt, b, nb);
        dim3 gB(nb * (C_IN * HW / 256));           // nb * 3136 blocks of 256
        invol_apply_kernel<<<gB, 256, 0, stream>>>(x, wgt, out, b, nb);
    }
}